// SpikeTruncatedMixtureModel_4355096838321
// MI455X (gfx1250) — compile-verified
//
#include <hip/hip_runtime.h>
#include <stdint.h>

// ---------------------------------------------------------------------------
// Truncated E-step for spike mixture model, CDNA5 / gfx1250.
//   n=262144 spikes, D=64, K=10 candidates, U=512 units, top-3 refresh.
// Memory-bound (~95MB @ 23.3TB/s => ~4us floor). CDNA5 paths used:
//   * TDM tensor_load_to_lds stages the 128KB mu matrix into LDS (TENSORcnt)
//   * V_WMMA_F32_16X16X4_F32 computes per-unit norms m2 = diag(Mu.Mu^T)
//   * ds_add_f32 LDS atomics accumulate per-WG m_u/N_u in a 512x64 LDS tile
//     (needs CDNA5's 320KB LDS/WGP), flushed once with global f32 atomics
//   * wave32 butterfly reductions, global_prefetch on the feature stream
// ---------------------------------------------------------------------------

#define K_CAND   10
#define D_DIM    64
#define U_UNITS  512
#define TOPK     3
#define BLOCK    256
#define WAVES    (BLOCK / 32)

typedef unsigned int u32x4 __attribute__((ext_vector_type(4)));
typedef int          i32x4 __attribute__((ext_vector_type(4)));
typedef int          i32x8 __attribute__((ext_vector_type(8)));
typedef float        v2f   __attribute__((ext_vector_type(2)));
typedef float        v8f   __attribute__((ext_vector_type(8)));

#if defined(__has_builtin)
#  if __has_builtin(__builtin_amdgcn_tensor_load_to_lds)
#    define HAVE_TDM 1
#  endif
#  if __has_builtin(__builtin_amdgcn_wmma_f32_16x16x4_f32)
#    define HAVE_WMMA 1
#  endif
#endif
#ifndef HAVE_TDM
#  define HAVE_TDM 0
#endif
#ifndef HAVE_WMMA
#  define HAVE_WMMA 0
#endif

__device__ __forceinline__ void wait_tensorcnt0() {
#if defined(__has_builtin) && __has_builtin(__builtin_amdgcn_s_wait_tensorcnt)
  __builtin_amdgcn_s_wait_tensorcnt(0);
#else
  asm volatile("s_wait_tensorcnt 0x0" ::: "memory");
#endif
}

#if HAVE_TDM
// Issue one TDM descriptor: load the whole U x D fp32 mu matrix (2-D tensor,
// tile == tensor) from global into LDS at byte offset lds_off.
__device__ __forceinline__ void tdm_load_mu(const float* mu, uint32_t lds_off) {
  uint64_t ga = (uint64_t)(uintptr_t)mu;
  u32x4 g0;
  g0.x = 1u;                                   // count=1, user mode, no gather
  g0.y = lds_off;                              // lds_addr (bytes)
  g0.z = (uint32_t)(ga & 0xFFFFFFFFu);         // global_addr[31:0]
  g0.w = (uint32_t)((ga >> 32) & 0x1FFFFFFu)   // global_addr[56:32]
       | (2u << 30);                           // type=2 ("image")
  // Group1 (256b): data_size=4B; tensor_dim0=64, tensor_dim1=512;
  // tile_dim0=64, tile_dim1=512; tensor_dim0_stride=64.
  i32x8 g1 = {
      (int)(2u << 16),                         // [17:16] data_size=2 (4B)
      (int)((uint32_t)D_DIM << 16),            // [63:48] tensor_dim0.lo16
      (int)((uint32_t)U_UNITS << 16),          // [95:80] tensor_dim1.lo16
      (int)((uint32_t)D_DIM << 16),            // [127:112] tile_dim0
      (int)U_UNITS,                            // [143:128] tile_dim1
      (int)D_DIM,                              // [191:160] dim0_stride.lo32
      0, 0};
  i32x4 z4 = {0, 0, 0, 0};
#if __clang_major__ >= 23
  i32x8 z8 = {0, 0, 0, 0, 0, 0, 0, 0};
  __builtin_amdgcn_tensor_load_to_lds(g0, g1, z4, z4, z8, 0);
#else
  __builtin_amdgcn_tensor_load_to_lds(g0, g1, z4, z4, 0);
#endif
}
#endif

__device__ __forceinline__ float wave_sum(float v) {
#pragma unroll
  for (int off = 16; off > 0; off >>= 1) v += __shfl_xor(v, off, 32);
  return v;
}

__global__ void __launch_bounds__(BLOCK) spike_zero_kernel(float* out, int count) {
  int i = blockIdx.x * blockDim.x + threadIdx.x;
  if (i < count) out[i] = 0.0f;
}

__global__ void __launch_bounds__(BLOCK) spike_estep_kernel(
    const float* __restrict__ features, const float* __restrict__ mu,
    const float* __restrict__ log_prop, const long long* __restrict__ cand,
    float* __restrict__ out, int n, int out_size) {
  extern __shared__ float smem[];
  float* s_mu   = smem;                       // 512*64 staged mu
  float* s_macc = s_mu + U_UNITS * D_DIM;     // 512*64 partial m_u
  float* s_nacc = s_macc + U_UNITS * D_DIM;   // 512 partial N_u
  float* s_m2   = s_nacc + U_UNITS;           // 512 |mu|^2
  float* s_lp   = s_m2 + U_UNITS;             // 512 log_prop
  float* s_red  = s_lp + U_UNITS;             // [elbo_sum, noise_N]

  const int tid  = threadIdx.x;
  const int wave = tid >> 5;
  const int lane = tid & 31;

  for (int i = tid; i < U_UNITS * D_DIM; i += BLOCK) s_macc[i] = 0.0f;
  for (int i = tid; i < U_UNITS; i += BLOCK) { s_nacc[i] = 0.0f; s_lp[i] = log_prop[i]; }
  if (tid < 2) s_red[tid] = 0.0f;

#if HAVE_TDM
  if (tid == 0) {
    // Low 32 bits of the generic LDS pointer == LDS byte offset (flat aperture).
    tdm_load_mu(mu, (uint32_t)(uintptr_t)s_mu);
    wait_tensorcnt0();
  }
#else
  for (int i = tid; i < U_UNITS * D_DIM; i += BLOCK) s_mu[i] = mu[i];
#endif
  __syncthreads();

#if HAVE_WMMA
  // m2 = diag(Mu.Mu^T) via V_WMMA_F32_16X16X4_F32 Gram tiles, 16 units/tile,
  // accumulating over D in K=4 chunks. A and B per-lane layouts coincide for
  // this symmetric case, so one LDS float2 load feeds both operands:
  //   lane L (L<16):  row=L,    holds K = d0+0, d0+1
  //   lane L (L>=16): row=L-16, holds K = d0+2, d0+3
  {
    const int row   = lane & 15;
    const int khalf = lane >> 4;
    for (int t = wave; t < U_UNITS / 16; t += WAVES) {
      const int u0 = t * 16;
      v8f c = {0.0f, 0.0f, 0.0f, 0.0f, 0.0f, 0.0f, 0.0f, 0.0f};
#pragma unroll
      for (int d0 = 0; d0 < D_DIM; d0 += 4) {
        const float2 m2l =
            *(const float2*)&s_mu[(u0 + row) * D_DIM + d0 + 2 * khalf];
        v2f a; a.x = m2l.x; a.y = m2l.y;
        c = __builtin_amdgcn_wmma_f32_16x16x4_f32(
            /*neg_a=*/false, a, /*neg_b=*/false, a,
            /*c_mod=*/(short)0, c, /*reuse_a=*/false, /*reuse_b=*/false);
      }
      // Diagonal M==N==row: lanes 0-15 hold M=0..7 (c[j] <-> M=j),
      // lanes 16-31 hold M=8..15 (c[j] <-> M=8+j).
      float diag = 0.0f;
      bool  have = false;
      if (lane < 16) {
        if (row < 8) {
          have = true;
#pragma unroll
          for (int j = 0; j < 8; ++j) if (j == row) diag = c[j];
        }
      } else {
        if (row >= 8) {
          have = true;
#pragma unroll
          for (int j = 0; j < 8; ++j) if (j == row - 8) diag = c[j];
        }
      }
      if (have) s_m2[u0 + row] = diag;
    }
  }
#else
  for (int u = tid; u < U_UNITS; u += BLOCK) {
    float acc = 0.0f;
    for (int d = 0; d < D_DIM; ++d) { float m = s_mu[u * D_DIM + d]; acc = fmaf(m, m, acc); }
    s_m2[u] = acc;
  }
#endif
  __syncthreads();

  const long long chunk = ((long long)n + gridDim.x - 1) / gridDim.x;
  const long long sBeg  = (long long)blockIdx.x * chunk;
  long long sEnd = sBeg + chunk; if (sEnd > n) sEnd = n;
  const int   topOff = 2 + U_UNITS + U_UNITS * D_DIM;        // 33282
  const bool  topI64 = (out_size - topOff) >= n * TOPK * 2;  // int64 layout?
  const float invn   = 1.0f / (float)n;

  float elboAcc = 0.0f, noiseAcc = 0.0f;

  for (long long s = sBeg + wave; s < sEnd; s += WAVES) {
    if (s + WAVES < sEnd) __builtin_prefetch(features + (s + WAVES) * D_DIM);

    const float2 f = ((const float2*)features)[s * (D_DIM / 2) + lane]; // 2 dims/lane
    const float x2 = wave_sum(f.x * f.x + f.y * f.y);

    float ll[K_CAND];
    int   unit[K_CAND];
#pragma unroll
    for (int k = 0; k < K_CAND; ++k) {
      const int u = (int)cand[s * K_CAND + k];   // wave-uniform
      unit[k] = u;
      const float2 m = ((const float2*)s_mu)[u * (D_DIM / 2) + lane];
      const float xm = wave_sum(f.x * m.x + f.y * m.y);
      ll[k] = -0.5f * (x2 - 2.0f * xm + s_m2[u]) + s_lp[u];
    }

    const float noise_ll = -0.5f * x2;
    float mx = noise_ll;
#pragma unroll
    for (int k = 0; k < K_CAND; ++k) mx = fmaxf(mx, ll[k]);
    float q[K_CAND];
    float se = __expf(noise_ll - mx);
#pragma unroll
    for (int k = 0; k < K_CAND; ++k) { q[k] = __expf(ll[k] - mx); se += q[k]; }
    const float inv = 1.0f / se;
    elboAcc  += mx + __logf(se);
    noiseAcc += __expf(noise_ll - mx) * inv;

    // Scatter sufficient stats into LDS (ds_add_f32; each lane owns 2 dims).
#pragma unroll
    for (int k = 0; k < K_CAND; ++k) {
      const float qk = q[k] * inv;
      float* dst = s_macc + unit[k] * D_DIM + 2 * lane;
      atomicAdd(dst, qk * f.x);
      atomicAdd(dst + 1, qk * f.y);
      if (lane == 0) atomicAdd(&s_nacc[unit[k]], qk);
    }

    // Top-3 of ll (ties -> lowest index, matching lax.top_k).
    int i0 = 0; float b0 = ll[0];
#pragma unroll
    for (int k = 1; k < K_CAND; ++k) if (ll[k] > b0) { b0 = ll[k]; i0 = k; }
    int i1 = 0; float b1 = -3.4e38f;
#pragma unroll
    for (int k = 0; k < K_CAND; ++k) if (k != i0 && ll[k] > b1) { b1 = ll[k]; i1 = k; }
    int i2 = 0; float b2 = -3.4e38f;
#pragma unroll
    for (int k = 0; k < K_CAND; ++k) if (k != i0 && k != i1 && ll[k] > b2) { b2 = ll[k]; i2 = k; }
    int c0 = 0, c1 = 0, c2 = 0;
#pragma unroll
    for (int k = 0; k < K_CAND; ++k) {
      if (k == i0) c0 = unit[k];
      if (k == i1) c1 = unit[k];
      if (k == i2) c2 = unit[k];
    }
    if (lane < TOPK) {
      const int sel = (lane == 0) ? c0 : ((lane == 1) ? c1 : c2);
      if (topI64) {
        ((long long*)(out + topOff))[s * TOPK + lane] = (long long)sel;
      } else {
        const long long idx = topOff + s * TOPK + lane;
        if (idx < out_size) out[idx] = (float)sel;
      }
    }
  }

  if (lane == 0) { atomicAdd(&s_red[0], elboAcc * invn); atomicAdd(&s_red[1], noiseAcc); }
  __syncthreads();

  // Flush per-WG partials to global with fp32 atomics.
  for (int i = tid; i < U_UNITS * D_DIM; i += BLOCK) atomicAdd(&out[2 + U_UNITS + i], s_macc[i]);
  for (int i = tid; i < U_UNITS; i += BLOCK) atomicAdd(&out[2 + i], s_nacc[i]);
  if (tid == 0) { atomicAdd(&out[0], s_red[0]); atomicAdd(&out[1], s_red[1]); }
}

extern "C" void kernel_launch(void* const* d_in, const int* in_sizes, int n_in,
                              void* d_out, int out_size, void* d_ws, size_t ws_size,
                              hipStream_t stream) {
  const float*     features = (const float*)d_in[0];
  const float*     mu       = (const float*)d_in[1];
  const float*     log_prop = (const float*)d_in[2];
  const long long* cand     = (const long long*)d_in[3];
  float*           out      = (float*)d_out;
  const int n = in_sizes[0] / D_DIM;

  // Zero the atomic accumulator region (scalars + N_u + m_u) every call.
  const int zc = 2 + U_UNITS + U_UNITS * D_DIM;
  spike_zero_kernel<<<(zc + BLOCK - 1) / BLOCK, BLOCK, 0, stream>>>(out, zc);

  const size_t smem = (size_t)(2 * U_UNITS * D_DIM + 3 * U_UNITS + 2) * sizeof(float);
  (void)hipFuncSetAttribute((const void*)spike_estep_kernel,
                            hipFuncAttributeMaxDynamicSharedMemorySize, (int)smem);
  int grid = (n + 511) / 512;
  if (grid < 1) grid = 1;
  spike_estep_kernel<<<grid, BLOCK, smem, stream>>>(features, mu, log_prop, cand,
                                                    out, n, out_size);
}